// GAT_51599737094368
// MI455X (gfx1250) — compile-verified
//
#include <hip/hip_runtime.h>
#include <cmath>

#define BLK 256
#define NEG_SLOPE 0.2f

typedef __attribute__((ext_vector_type(2))) float v2f;
typedef __attribute__((ext_vector_type(8))) float v8f;

// ---------- order-preserving float<->uint for atomic segment-max ----------
__device__ __forceinline__ unsigned ordf(float f) {
  unsigned u = __float_as_uint(f);
  return (u & 0x80000000u) ? ~u : (u | 0x80000000u);
}
__device__ __forceinline__ float unordf(unsigned u) {
  return __uint_as_float((u & 0x80000000u) ? (u & 0x7fffffffu) : ~u);
}
#define ORD_NEG_INF 0x007FFFFFu   // ordf(-inf)

__device__ __forceinline__ void edge_sd(const long long* __restrict__ ei,
                                        int e, int E, int& s, int& d) {
  if (e < E) { s = (int)ei[e]; d = (int)ei[(size_t)E + e]; }
  else       { s = e - E; d = s; }   // appended self-loop
}

// ---------------------------- fill ----------------------------
__global__ void fill_u32_kernel(unsigned* __restrict__ p, unsigned v, size_t n) {
  size_t i = (size_t)blockIdx.x * blockDim.x + threadIdx.x;
  if (i < n) p[i] = v;
}

// ---------------------------- fp32 WMMA GEMM ----------------------------
// C[M x (NT*16)] = A[M x K] * B[K x ldb], row-major. Each wave: 16 rows, NT col tiles.
// V_WMMA_F32_16X16X4_F32 fragment layout (ISA 7.12.2):
//   A 16x4 : lane&15 = M row; VGPR{0,1} = K{0,1} (lanes 0-15) / K{2,3} (lanes 16-31)
//   B 4x16 : lane&15 = N col; same K split per lane-half
//   C/D    : lane&15 = N col; VGPR v = row v (lanes 0-15) / row v+8 (lanes 16-31)
template <int NT>
__global__ __launch_bounds__(BLK) void wmma_gemm_f32(
    const float* __restrict__ A, const float* __restrict__ B,
    float* __restrict__ C, int M, int K, int ldb, int ldc) {
  int wave = (int)((blockIdx.x * (size_t)blockDim.x + threadIdx.x) >> 5);
  int lane = threadIdx.x & 31;
  int row0 = wave * 16;
  if (row0 >= M) return;

  int r      = row0 + (lane & 15);
  int khalf  = (lane >> 4) * 2;        // 0 or 2
  int col    = lane & 15;

  v8f acc[NT] = {};
  for (int k0 = 0; k0 < K; k0 += 4) {
    int kb = k0 + khalf;
    const float* ap = A + (size_t)r * K + kb;
    v2f a; a.x = ap[0]; a.y = ap[1];
#pragma unroll
    for (int t = 0; t < NT; ++t) {
      const float* bp = B + (size_t)kb * ldb + t * 16 + col;
      v2f b; b.x = bp[0]; b.y = bp[ldb];
      acc[t] = __builtin_amdgcn_wmma_f32_16x16x4_f32(
          false, a, false, b, (short)0, acc[t], false, false);
    }
  }
  int mrow = row0 + (lane >> 4) * 8;
#pragma unroll
  for (int t = 0; t < NT; ++t) {
#pragma unroll
    for (int v = 0; v < 8; ++v)
      C[(size_t)(mrow + v) * ldc + t * 16 + col] = acc[t][v];
  }
}

// ------------------- edge logits + segment max -------------------
// one thread per (edge, head): logit = sum_d lrelu(xl[s]+xr[d]) * att
__global__ void gat_logits_kernel(const float* __restrict__ xl,
                                  const float* __restrict__ xr,
                                  const float* __restrict__ att,
                                  const long long* __restrict__ ei,
                                  int E, int E2, int H, int DH, int LS,
                                  float* __restrict__ logits,
                                  unsigned* __restrict__ mord) {
  int i = blockIdx.x * blockDim.x + threadIdx.x;
  if (i >= E2 * H) return;
  int e = i / H, h = i - e * H;
  int s, d; edge_sd(ei, e, E, s, d);
  const float* pl = xl + (size_t)s * LS + h * DH;
  const float* pr = xr + (size_t)d * LS + h * DH;
  const float* pa = att + h * DH;
  float acc = 0.f;
  for (int k = 0; k < DH; ++k) {
    float v = pl[k] + pr[k];
    v = v > 0.f ? v : NEG_SLOPE * v;
    acc = fmaf(v, pa[k], acc);
  }
  logits[i] = acc;
  atomicMax(&mord[(size_t)d * H + h], ordf(acc));
}

// ------------------- exp(logit - max) + segment sum -------------------
__global__ void gat_exp_kernel(const long long* __restrict__ ei, int E, int E2,
                               int H, const unsigned* __restrict__ mord,
                               float* __restrict__ ex, float* __restrict__ denom) {
  int i = blockIdx.x * blockDim.x + threadIdx.x;
  if (i >= E2 * H) return;
  int e = i / H, h = i - e * H;
  int s, d; edge_sd(ei, e, E, s, d); (void)s;
  float m = unordf(mord[(size_t)d * H + h]);
  float v = expf(ex[i] - m);
  ex[i] = v;
  atomicAdd(&denom[(size_t)d * H + h], v);
}

// ------------------- weighted scatter: agg[d] += alpha * xl[s] -------------------
__global__ void gat_scatter_kernel(const long long* __restrict__ ei, int E, int E2,
                                   int H, int DH, int FD, int LS, int total,
                                   const float* __restrict__ ex,
                                   const float* __restrict__ denom,
                                   const float* __restrict__ xl,
                                   float* __restrict__ agg) {
  int i = blockIdx.x * blockDim.x + threadIdx.x;
  if (i >= total) return;
  int e = i / FD, c = i - e * FD, h = c / DH;
  int s, d; edge_sd(ei, e, E, s, d);
  float alpha = ex[(size_t)e * H + h] / (denom[(size_t)d * H + h] + 1e-16f);
  atomicAdd(&agg[(size_t)d * FD + c], alpha * xl[(size_t)s * LS + c]);
}

// ------------------- bias + ELU (in place) -------------------
__global__ void elu_bias_kernel(float* __restrict__ h, const float* __restrict__ b,
                                int FD, size_t n) {
  size_t i = (size_t)blockIdx.x * blockDim.x + threadIdx.x;
  if (i >= n) return;
  float v = h[i] + b[i % FD];
  h[i] = v > 0.f ? v : expf(v) - 1.f;
}

// ------------------- pad W2 [128 x 40] -> [128 x 48] -------------------
__global__ void pad_w_kernel(const float* __restrict__ W, float* __restrict__ Wp,
                             int rows, int cin, int cout) {
  int i = blockIdx.x * blockDim.x + threadIdx.x;
  if (i >= rows * cout) return;
  int r = i / cout, c = i - r * cout;
  Wp[i] = (c < cin) ? W[(size_t)r * cin + c] : 0.f;
}

// ------------------- bias + log_softmax over 40 -------------------
__global__ void logsoftmax_kernel(const float* __restrict__ agg,
                                  const float* __restrict__ b,
                                  float* __restrict__ out, int N) {
  int n = blockIdx.x * blockDim.x + threadIdx.x;
  if (n >= N) return;
  float v[40];
  float m = -INFINITY;
#pragma unroll
  for (int c = 0; c < 40; ++c) { v[c] = agg[(size_t)n * 40 + c] + b[c]; m = fmaxf(m, v[c]); }
  float s = 0.f;
#pragma unroll
  for (int c = 0; c < 40; ++c) s += expf(v[c] - m);
  float lse = logf(s);
#pragma unroll
  for (int c = 0; c < 40; ++c) out[(size_t)n * 40 + c] = v[c] - m - lse;
}

// =============================== host ===============================
static inline unsigned nblk(long long n, int b) { return (unsigned)((n + b - 1) / b); }

extern "C" void kernel_launch(void* const* d_in, const int* in_sizes, int n_in,
                              void* d_out, int out_size, void* d_ws, size_t ws_size,
                              hipStream_t stream) {
  const float*     x    = (const float*)d_in[0];
  const long long* ei   = (const long long*)d_in[1];
  const float*     Wl1  = (const float*)d_in[2];
  const float*     Wr1  = (const float*)d_in[3];
  const float*     att1 = (const float*)d_in[4];
  const float*     b1   = (const float*)d_in[5];
  const float*     Wl2  = (const float*)d_in[6];
  const float*     Wr2  = (const float*)d_in[7];
  const float*     att2 = (const float*)d_in[8];
  const float*     b2   = (const float*)d_in[9];
  float*           out  = (float*)d_out;

  const int N  = in_sizes[0] / 128;   // 50000
  const int E  = in_sizes[1] / 2;     // 800000
  const int E2 = E + N;               // with self-loops
  const size_t Nf = (size_t)N;

  // ---- workspace layout (floats); layer-2 overlays xr1/ex1/m1/den1 ----
  float* w = (float*)d_ws;
  float*    xl1  = w;                                 // N*128
  float*    xr1  = xl1 + Nf * 128;                    // N*128
  float*    agg1 = xr1 + Nf * 128;                    // N*128 (becomes h1)
  float*    ex1  = agg1 + Nf * 128;                   // E2*8
  unsigned* m1   = (unsigned*)(ex1 + (size_t)E2 * 8); // N*8
  float*    den1 = (float*)(m1 + Nf * 8);             // N*8
  // layer-2 overlays (valid after layer-1 scatter finishes; same stream => ordered)
  float*    xl2  = xr1;                               // N*48
  float*    xr2  = xl2 + Nf * 48;                     // N*48 (fits in xr1's N*128)
  float*    ex2  = ex1;                               // E2
  unsigned* m2   = (unsigned*)(ex2 + (size_t)E2);     // N
  float*    den2 = (float*)(m2 + Nf);                 // N
  float*    out2 = den2 + Nf;                         // N*40
  float*    w2l  = out2 + Nf * 40;                    // 128*48
  float*    w2r  = w2l + 128 * 48;                    // 128*48

  const int gemm_waves = (N + 15) / 16;
  const unsigned gemm_blocks = nblk((long long)gemm_waves * 32, BLK);

  // =================== Layer 1 (H=8, DH=16, F=128) ===================
  fill_u32_kernel<<<nblk(Nf * 8, BLK), BLK, 0, stream>>>(m1, ORD_NEG_INF, Nf * 8);
  fill_u32_kernel<<<nblk(Nf * 8, BLK), BLK, 0, stream>>>((unsigned*)den1, 0u, Nf * 8);
  fill_u32_kernel<<<nblk(Nf * 128, BLK), BLK, 0, stream>>>((unsigned*)agg1, 0u, Nf * 128);

  wmma_gemm_f32<8><<<gemm_blocks, BLK, 0, stream>>>(x, Wl1, xl1, N, 128, 128, 128);
  wmma_gemm_f32<8><<<gemm_blocks, BLK, 0, stream>>>(x, Wr1, xr1, N, 128, 128, 128);

  gat_logits_kernel<<<nblk((long long)E2 * 8, BLK), BLK, 0, stream>>>(
      xl1, xr1, att1, ei, E, E2, 8, 16, 128, ex1, m1);
  gat_exp_kernel<<<nblk((long long)E2 * 8, BLK), BLK, 0, stream>>>(
      ei, E, E2, 8, m1, ex1, den1);
  gat_scatter_kernel<<<nblk((long long)E2 * 128, BLK), BLK, 0, stream>>>(
      ei, E, E2, 8, 16, 128, 128, E2 * 128, ex1, den1, xl1, agg1);
  elu_bias_kernel<<<nblk(Nf * 128, BLK), BLK, 0, stream>>>(agg1, b1, 128, Nf * 128);

  // =================== Layer 2 (H=1, D=40 padded to 48) ===================
  pad_w_kernel<<<nblk(128 * 48, BLK), BLK, 0, stream>>>(Wl2, w2l, 128, 40, 48);
  pad_w_kernel<<<nblk(128 * 48, BLK), BLK, 0, stream>>>(Wr2, w2r, 128, 40, 48);
  fill_u32_kernel<<<nblk(Nf, BLK), BLK, 0, stream>>>(m2, ORD_NEG_INF, Nf);
  fill_u32_kernel<<<nblk(Nf, BLK), BLK, 0, stream>>>((unsigned*)den2, 0u, Nf);
  fill_u32_kernel<<<nblk(Nf * 40, BLK), BLK, 0, stream>>>((unsigned*)out2, 0u, Nf * 40);

  wmma_gemm_f32<3><<<gemm_blocks, BLK, 0, stream>>>(agg1, w2l, xl2, N, 128, 48, 48);
  wmma_gemm_f32<3><<<gemm_blocks, BLK, 0, stream>>>(agg1, w2r, xr2, N, 128, 48, 48);

  gat_logits_kernel<<<nblk((long long)E2, BLK), BLK, 0, stream>>>(
      xl2, xr2, att2, ei, E, E2, 1, 40, 48, ex2, m2);
  gat_exp_kernel<<<nblk((long long)E2, BLK), BLK, 0, stream>>>(
      ei, E, E2, 1, m2, ex2, den2);
  gat_scatter_kernel<<<nblk((long long)E2 * 40, BLK), BLK, 0, stream>>>(
      ei, E, E2, 1, 40, 40, 48, E2 * 40, ex2, den2, xl2, out2);

  logsoftmax_kernel<<<nblk(Nf, BLK), BLK, 0, stream>>>(out2, b2, out, N);
}